// RbfKernel_65481071397592
// MI455X (gfx1250) — compile-verified
//
#include <hip/hip_runtime.h>

// RBF kernel exp(-gamma * ||x_i - y_j||^2) on MI455X (gfx1250).
// Compute path: V_WMMA_F32_16X16X4_F32 (full-precision f32 matrix core op).
// The problem is bound by the 256 MB output stream (~11us at 23.3 TB/s HBM),
// so fp32 WMMA is chosen for exact-matching numerics at zero cost.

typedef __attribute__((ext_vector_type(2))) float v2f;
typedef __attribute__((ext_vector_type(8))) float v8f;

#define NROWS 8192
#define DDIM  64
#define GAMMA 1.0f

// ---------------------------------------------------------------------------
// Prologue: per-row squared norms of x and y into workspace (2 * 8192 floats).
// ---------------------------------------------------------------------------
__global__ void __launch_bounds__(256)
rbf_row_norms(const float* __restrict__ x, const float* __restrict__ y,
              float* __restrict__ x2, float* __restrict__ y2) {
    int row = blockIdx.x * blockDim.x + threadIdx.x;
    if (row >= NROWS) return;
    const float4* xr = (const float4*)(x + (size_t)row * DDIM);
    const float4* yr = (const float4*)(y + (size_t)row * DDIM);
    float sx = 0.0f, sy = 0.0f;
#pragma unroll
    for (int i = 0; i < DDIM / 4; ++i) {
        float4 a = xr[i];
        sx += a.x * a.x + a.y * a.y + a.z * a.z + a.w * a.w;
        float4 b = yr[i];
        sy += b.x * b.x + b.y * b.y + b.z * b.z + b.w * b.w;
    }
    x2[row] = sx;
    y2[row] = sy;
}

// ---------------------------------------------------------------------------
// Main kernel: one wave32 per 16x16 output tile.
//   A-fragment (16x4 f32): lane l (half h = l>>4) holds x[rowBase + (l&15)][k + 2h .. +1]
//   B-fragment (4x16 f32): mirrored layout on y rows (acting as columns of y^T)
//   C/D (16x16 f32, 8 VGPRs): VGPR r of lane l = element (M = r + 8h, N = l&15)
// K = 64 -> 16 chained v_wmma_f32_16x16x4_f32 per tile. Epilogue fuses the
// distance assembly + exp and streams the result with non-temporal stores.
// ---------------------------------------------------------------------------
__global__ void __launch_bounds__(256)
rbf_wmma(const float* __restrict__ x, const float* __restrict__ y,
         const float* __restrict__ x2, const float* __restrict__ y2,
         float* __restrict__ out) {
    const int lane = threadIdx.x;        // 0..31 (wave32)
    const int half = lane >> 4;          // 0 or 1
    const int lrow = lane & 15;

    // 8 waves per block; consecutive waves -> consecutive column tiles
    // (same row tile of x stays hot in L2/L0).
    const int tile = blockIdx.x * 8 + threadIdx.y;
    const int trow = tile >> 9;          // 8192/16 = 512 tiles per dim
    const int tcol = tile & 511;
    const int rowBase = trow * 16;
    const int colBase = tcol * 16;

    const float* xa = x + (size_t)(rowBase + lrow) * DDIM + 2 * half;
    const float* yb = y + (size_t)(colBase + lrow) * DDIM + 2 * half;

    v8f acc = {};
#pragma unroll
    for (int ks = 0; ks < DDIM / 4; ++ks) {
        v2f afrag = *(const v2f*)(xa + ks * 4);
        v2f bfrag = *(const v2f*)(yb + ks * 4);
        // (neg_a, A, neg_b, B, c_mod, C, reuse_a, reuse_b)
        acc = __builtin_amdgcn_wmma_f32_16x16x4_f32(
            false, afrag, false, bfrag, (short)0, acc, false, false);
    }

    // Epilogue: dist2 = max(x2 - 2*xy + y2, 0); out = exp(-gamma*dist2).
    const float ycol = y2[colBase + lrow];
    float* orow = out + (size_t)(rowBase + 8 * half) * NROWS + colBase + lrow;
#pragma unroll
    for (int r = 0; r < 8; ++r) {
        float xr2 = x2[rowBase + 8 * half + r];
        float d2  = fmaxf(xr2 - 2.0f * acc[r] + ycol, 0.0f);
        float v   = __expf(-GAMMA * d2);
        // 256 MB streaming output > 192 MB L2: bypass with non-temporal store.
        __builtin_nontemporal_store(v, orow + (size_t)r * NROWS);
    }
}

// ---------------------------------------------------------------------------
extern "C" void kernel_launch(void* const* d_in, const int* in_sizes, int n_in,
                              void* d_out, int out_size, void* d_ws, size_t ws_size,
                              hipStream_t stream) {
    const float* x = (const float*)d_in[0];
    const float* y = (const float*)d_in[1];
    float* out = (float*)d_out;

    float* x2 = (float*)d_ws;            // 8192 floats
    float* y2 = x2 + NROWS;              // 8192 floats (64 KB total)

    rbf_row_norms<<<NROWS / 256, 256, 0, stream>>>(x, y, x2, y2);

    const int tiles = (NROWS / 16) * (NROWS / 16);   // 262144 tiles
    dim3 block(32, 8);                               // 8 waves / block
    rbf_wmma<<<tiles / 8, block, 0, stream>>>(x, y, x2, y2, out);
}